// NoisyPositionEmbedder_58222576664942
// MI455X (gfx1250) — compile-verified
//
#include <hip/hip_runtime.h>
#include <hip/hip_bf16.h>

typedef __attribute__((ext_vector_type(2))) float v2f;
typedef __attribute__((ext_vector_type(8))) float v8f;

#define N_ATOM   1024
#define N_TOKEN  256
#define C_S      384
#define C_Zc     128
#define C_ATOM   128
#define C_AP     16
#define LN_EPS   1e-5f

// ---------------------------------------------------------------------------
// Kernel A: recover token index per atom from the one-hot map.
// ---------------------------------------------------------------------------
__global__ __launch_bounds__(256) void tok_kernel(const float* __restrict__ a2t,
                                                  int* __restrict__ tok) {
    int a = blockIdx.x * 256 + threadIdx.x;
    if (a >= N_ATOM) return;
    const float* row = a2t + (size_t)a * N_TOKEN;
    int t = 0;
    for (int j = 0; j < N_TOKEN; ++j)
        if (row[j] > 0.5f) t = j;
    tok[a] = t;
}

// ---------------------------------------------------------------------------
// Kernel B: u[t] = LN_s(s_trunk[t]) @ W_s   (256 tokens, 384 -> 128)
// One block (128 threads) per token.
// ---------------------------------------------------------------------------
__global__ __launch_bounds__(128) void token_s_kernel(
    const float* __restrict__ s_trunk, const float* __restrict__ lnw,
    const float* __restrict__ lnb, const float* __restrict__ Ws,
    float* __restrict__ u) {
    __shared__ float xn[C_S];
    __shared__ float red[128];
    const int t = blockIdx.x;
    const int tid = threadIdx.x;
    const float* srow = s_trunk + (size_t)t * C_S;
    float a0 = srow[tid], a1 = srow[tid + 128], a2 = srow[tid + 256];

    red[tid] = a0 + a1 + a2;
    __syncthreads();
    for (int off = 64; off > 0; off >>= 1) {
        if (tid < off) red[tid] += red[tid + off];
        __syncthreads();
    }
    const float mean = red[0] * (1.f / (float)C_S);
    __syncthreads();

    float d0 = a0 - mean, d1 = a1 - mean, d2 = a2 - mean;
    red[tid] = d0 * d0 + d1 * d1 + d2 * d2;
    __syncthreads();
    for (int off = 64; off > 0; off >>= 1) {
        if (tid < off) red[tid] += red[tid + off];
        __syncthreads();
    }
    const float rstd = rsqrtf(red[0] * (1.f / (float)C_S) + LN_EPS);
    __syncthreads();

    xn[tid]       = d0 * rstd * lnw[tid]       + lnb[tid];
    xn[tid + 128] = d1 * rstd * lnw[tid + 128] + lnb[tid + 128];
    xn[tid + 256] = d2 * rstd * lnw[tid + 256] + lnb[tid + 256];
    __syncthreads();

    float acc = 0.f;
    #pragma unroll 4
    for (int c = 0; c < C_S; ++c)
        acc = fmaf(xn[c], Ws[(size_t)c * C_ATOM + tid], acc);
    u[(size_t)t * C_ATOM + tid] = acc;
}

// ---------------------------------------------------------------------------
// Kernel C: y[t,m] = LN_z(zij[t,m]) @ W_z for all 65536 token pairs.
// One wave32 per tile of 16 pairs; K=128 via 32x V_WMMA_F32_16X16X4_F32.
// A-fragment layout (16x4 f32): lanes 0-15 hold K offsets {0,1}, lanes 16-31
// hold {2,3}, so each lane loads only cols = {koff, koff+1} mod 4 of its row.
// ---------------------------------------------------------------------------
__global__ __launch_bounds__(128) void pair_ln_wz_kernel(
    const float* __restrict__ zij, const float* __restrict__ lnw,
    const float* __restrict__ lnb, const float* __restrict__ Wz,
    float* __restrict__ y) {
    __shared__ float sWz[C_Zc * C_AP];  // 8 KB
    for (int i = threadIdx.x; i < C_Zc * C_AP; i += 128) sWz[i] = Wz[i];
    __syncthreads();

    const int lane = threadIdx.x & 31;
    const int wave = threadIdx.x >> 5;
    const int tile = blockIdx.x * 4 + wave;       // 4096 tiles total
    const int p0   = tile * 16;                   // first token-pair of tile
    const int row  = lane & 15;
    const int koff = (lane < 16) ? 0 : 2;
    const int n    = lane & 15;                   // output channel (B/D layout)
    const float* zrow = zij + (size_t)(p0 + row) * C_Zc;

    // Load this lane's 64 elements (32 float2) + LN statistics.
    float2 xv[32];
    float s = 0.f, s2 = 0.f;
    #pragma unroll
    for (int j = 0; j < 32; ++j) {
        float2 v = *(const float2*)(zrow + 4 * j + koff);
        xv[j] = v;
        s  += v.x + v.y;
        s2 += v.x * v.x + v.y * v.y;
    }
    s  += __shfl_xor(s,  16, 32);   // combine the two column-halves of the row
    s2 += __shfl_xor(s2, 16, 32);
    const float mean = s * (1.f / (float)C_Zc);
    const float var  = s2 * (1.f / (float)C_Zc) - mean * mean;
    const float rstd = rsqrtf(var + LN_EPS);

    // Normalize + scale/shift in registers.
    #pragma unroll
    for (int j = 0; j < 32; ++j) {
        const int c = 4 * j + koff;
        float2 w = *(const float2*)(lnw + c);
        float2 b = *(const float2*)(lnb + c);
        xv[j].x = (xv[j].x - mean) * rstd * w.x + b.x;
        xv[j].y = (xv[j].y - mean) * rstd * w.y + b.y;
    }

    // 32 chained f32 WMMAs: D(16x16) = A(16x128) * Wz(128x16)
    v8f acc = {0.f, 0.f, 0.f, 0.f, 0.f, 0.f, 0.f, 0.f};
    #pragma unroll
    for (int kb = 0; kb < 32; ++kb) {
        v2f a; a.x = xv[kb].x; a.y = xv[kb].y;
        const int kr = 4 * kb + koff;
        v2f b; b.x = sWz[kr * C_AP + n]; b.y = sWz[(kr + 1) * C_AP + n];
        acc = __builtin_amdgcn_wmma_f32_16x16x4_f32(
            false, a, false, b, (short)0, acc, false, false);
    }

    // D layout: VGPR v -> pair row (v) lanes 0-15 / (v+8) lanes 16-31, col n.
    const int rbase = p0 + ((lane < 16) ? 0 : 8);
    #pragma unroll
    for (int v = 0; v < 8; ++v)
        y[(size_t)(rbase + v) * C_AP + n] = acc[v];
}

// ---------------------------------------------------------------------------
// Kernel D1: cl_out = cl + u[tok[a]] ; ql_out = ql + rl @ W_r  (fused)
// ---------------------------------------------------------------------------
__global__ __launch_bounds__(256) void cl_ql_kernel(
    const float* __restrict__ cl, const float* __restrict__ ql,
    const float* __restrict__ rl, const float* __restrict__ Wr,
    const float* __restrict__ u, const int* __restrict__ tok,
    float* __restrict__ cl_out, float* __restrict__ ql_out) {
    const int i = blockIdx.x * 256 + threadIdx.x;   // 131072 total
    const int a = i >> 7;
    const int c = i & 127;
    cl_out[i] = cl[i] + u[(size_t)tok[a] * C_ATOM + c];
    const float r0 = rl[a * 3 + 0], r1 = rl[a * 3 + 1], r2 = rl[a * 3 + 2];
    ql_out[i] = ql[i] + r0 * Wr[c] + r1 * Wr[128 + c] + r2 * Wr[256 + c];
}

// ---------------------------------------------------------------------------
// Kernel D2: plm_out[a,b,:] = plm[a,b,:] + y[tok[a],tok[b],:]
// Pure HBM stream (134 MB) with float4 accesses; y is L2-resident (4 MB).
// ---------------------------------------------------------------------------
__global__ __launch_bounds__(256) void plm_kernel(
    const float4* __restrict__ plm, const float4* __restrict__ y4,
    const int* __restrict__ tok, float4* __restrict__ out) {
    const size_t i = (size_t)blockIdx.x * 256 + threadIdx.x;  // 4194304 total
    const int pair = (int)(i >> 2);
    const int q    = (int)(i & 3);
    const int a = pair >> 10;
    const int b = pair & 1023;
    const int tp = tok[a] * N_TOKEN + tok[b];
    const float4 pv = plm[i];
    const float4 yv = y4[(size_t)tp * 4 + q];
    float4 o;
    o.x = pv.x + yv.x; o.y = pv.y + yv.y;
    o.z = pv.z + yv.z; o.w = pv.w + yv.w;
    out[i] = o;
}

// ---------------------------------------------------------------------------
extern "C" void kernel_launch(void* const* d_in, const int* in_sizes, int n_in,
                              void* d_out, int out_size, void* d_ws, size_t ws_size,
                              hipStream_t stream) {
    (void)in_sizes; (void)n_in; (void)out_size; (void)ws_size;

    const float* a2t     = (const float*)d_in[0];
    const float* cl      = (const float*)d_in[1];
    const float* plm     = (const float*)d_in[2];
    const float* ql      = (const float*)d_in[3];
    const float* s_trunk = (const float*)d_in[4];
    const float* zij     = (const float*)d_in[5];
    const float* rl      = (const float*)d_in[6];
    const float* ln_s_w  = (const float*)d_in[7];
    const float* ln_s_b  = (const float*)d_in[8];
    const float* W_s     = (const float*)d_in[9];
    const float* ln_z_w  = (const float*)d_in[10];
    const float* ln_z_b  = (const float*)d_in[11];
    const float* W_z     = (const float*)d_in[12];
    const float* W_r     = (const float*)d_in[13];

    // Workspace layout (bytes): tok[1024 i32] | u[256*128 f32] | y[65536*16 f32]
    char* ws = (char*)d_ws;
    int*   tok = (int*)ws;                                   // 4 KB
    float* u   = (float*)(ws + 4096);                        // 128 KB
    float* y   = (float*)(ws + 4096 + (size_t)N_TOKEN * C_ATOM * 4);  // 4 MB

    // Output layout: cl | plm | ql
    float* cl_out  = (float*)d_out;
    float* plm_out = cl_out + (size_t)N_ATOM * C_ATOM;
    float* ql_out  = plm_out + (size_t)N_ATOM * N_ATOM * C_AP;

    tok_kernel<<<(N_ATOM + 255) / 256, 256, 0, stream>>>(a2t, tok);

    token_s_kernel<<<N_TOKEN, 128, 0, stream>>>(s_trunk, ln_s_w, ln_s_b, W_s, u);

    // 65536 token pairs / (16 pairs per wave * 4 waves per block) = 1024 blocks
    pair_ln_wz_kernel<<<(N_TOKEN * N_TOKEN) / 64, 128, 0, stream>>>(
        zij, ln_z_w, ln_z_b, W_z, y);

    cl_ql_kernel<<<(N_ATOM * C_ATOM) / 256, 256, 0, stream>>>(
        cl, ql, rl, W_r, u, tok, cl_out, ql_out);

    plm_kernel<<<(int)(((size_t)N_ATOM * N_ATOM * C_AP / 4) / 256), 256, 0, stream>>>(
        (const float4*)plm, (const float4*)y, tok, (float4*)plm_out);
}